// JKE_83468394431312
// MI455X (gfx1250) — compile-verified
//
#include <hip/hip_runtime.h>
#include <math.h>

typedef __attribute__((ext_vector_type(2))) float v2f;
typedef __attribute__((ext_vector_type(8))) float v8f;

#define NN 20000
#define DD 64
#define HH 512
#define EE 320000

// ---------------- degree: deg[dst] += 1 ----------------
__global__ void deg_kernel(const int* __restrict__ dst, float* __restrict__ deg, int E) {
    int i = blockIdx.x * blockDim.x + threadIdx.x;
    if (i < E) atomicAdd(&deg[dst[i]], 1.0f);
}

// ------- scatter-add: agg[dst] += x[src] * ew, float4 per thread -------
__global__ void scatter_kernel(const float* __restrict__ x, const int* __restrict__ src,
                               const int* __restrict__ dst, const float* __restrict__ ew,
                               float* __restrict__ agg, int E, int F) {
    int per = F >> 2;                       // float4 chunks per row
    long long total = (long long)E * per;
    long long i = (long long)blockIdx.x * blockDim.x + threadIdx.x;
    if (i >= total) return;
    int e = (int)(i / per);
    int j = (int)(i - (long long)e * per) << 2;
    int s = src[e], d = dst[e];
    float w = ew ? ew[e] : 1.0f;
    float4 v = *(const float4*)(x + (long long)s * F + j);
    float* ap = agg + (long long)d * F + j;
    atomicAdd(ap + 0, v.x * w);
    atomicAdd(ap + 1, v.y * w);
    atomicAdd(ap + 2, v.z * w);
    atomicAdd(ap + 3, v.w * w);
}

// ---------------- agg /= max(deg, 1) ----------------
__global__ void normalize_kernel(float* __restrict__ agg, const float* __restrict__ deg,
                                 int n, int F) {
    int i = blockIdx.x * blockDim.x + threadIdx.x;
    if (i >= n * F) return;
    float dg = deg[i / F];
    agg[i] *= 1.0f / fmaxf(dg, 1.0f);
}

// --------- out = act(X @ W1 [+ A @ W2] + bias) via V_WMMA_F32_16X16X4_F32 ---------
// Block = 4 wave32s, all working on ONE output-tile column (tn = blockIdx.x).
// The K x 16 weight panels for this column are staged in LDS once, then each
// wave sweeps M-tiles (grid-stride over blockIdx.y/threadIdx.y), reading B
// fragments from LDS (conflict-free: lanes 0-15 and 16-31 hit disjoint
// 16-bank groups) and A fragments from global as b64 loads.
// M, K, Nout are multiples of 16/4 -> no tails, EXEC all-1s at every WMMA.
// ACT: 0=none, 1=relu, 2=tanh.
template <bool DUAL, int ACT>
__global__ void gemm_panel_wmma(const float* __restrict__ X, const float* __restrict__ W1,
                                const float* __restrict__ A, const float* __restrict__ W2,
                                const float* __restrict__ bias, float* __restrict__ out,
                                int M, int K, int Nout) {
    extern __shared__ float smem[];
    float* ldsW1 = smem;                       // K*16 floats
    float* ldsW2 = smem + K * 16;              // K*16 floats (DUAL only)

    const int tn      = blockIdx.x;
    const int colBase = tn * 16;
    const int nthr    = blockDim.x * blockDim.y;
    const int tid     = threadIdx.y * blockDim.x + threadIdx.x;

    // cooperative panel load: LDS layout [k][16]
    for (int idx = tid; idx < K * 16; idx += nthr) {
        int k = idx >> 4, n = idx & 15;
        ldsW1[idx] = W1[(long long)k * Nout + colBase + n];
        if (DUAL) ldsW2[idx] = W2[(long long)k * Nout + colBase + n];
    }
    __syncthreads();

    const int lane = threadIdx.x;              // 0..31
    const int r    = lane & 15;                // M (A-frag) / N (B-frag, store)
    const int koff = (lane >> 4) << 1;         // 0 (lanes 0-15) / 2 (lanes 16-31)
    const int col  = colBase + r;
    const float bv = bias ? bias[col] : 0.0f;

    const int Mtiles = M >> 4;
    const int step   = gridDim.y * blockDim.y;

    for (int tm = blockIdx.y * blockDim.y + threadIdx.y; tm < Mtiles; tm += step) {
        const float* xrow = X + (long long)(tm * 16 + r) * K + koff;
        const float* arow = DUAL ? (A + (long long)(tm * 16 + r) * K + koff) : nullptr;
        v8f acc = {};
        #pragma unroll 4
        for (int k0 = 0; k0 < K; k0 += 4) {
            v2f a1 = *(const v2f*)(xrow + k0);
            v2f b1;
            b1.x = ldsW1[(k0 + koff)     * 16 + r];
            b1.y = ldsW1[(k0 + koff + 1) * 16 + r];
            acc = __builtin_amdgcn_wmma_f32_16x16x4_f32(false, a1, false, b1,
                                                        (short)0, acc, false, false);
            if (DUAL) {
                v2f a2 = *(const v2f*)(arow + k0);
                v2f b2;
                b2.x = ldsW2[(k0 + koff)     * 16 + r];
                b2.y = ldsW2[(k0 + koff + 1) * 16 + r];
                acc = __builtin_amdgcn_wmma_f32_16x16x4_f32(false, a2, false, b2,
                                                            (short)0, acc, false, false);
            }
        }
        const int rbase = tm * 16 + ((lane >> 4) << 3);   // +0 / +8
        #pragma unroll
        for (int i = 0; i < 8; ++i) {
            float v = acc[i] + bv;
            if (ACT == 1)      v = fmaxf(v, 0.0f);
            else if (ACT == 2) v = tanhf(v);
            out[(long long)(rbase + i) * Nout + col] = v;
        }
    }
}

// ---------------- sum_i a[i]*b[i]  -> atomicAdd(accum) ----------------
__global__ void dot_reduce_kernel(const float* __restrict__ a, const float* __restrict__ b,
                                  float* __restrict__ accum, int n) {
    __shared__ float sdata[256];
    float v = 0.0f;
    for (long long idx = (long long)blockIdx.x * blockDim.x + threadIdx.x; idx < n;
         idx += (long long)gridDim.x * blockDim.x)
        v += a[idx] * b[idx];
    sdata[threadIdx.x] = v;
    __syncthreads();
    for (int s = blockDim.x >> 1; s > 0; s >>= 1) {
        if (threadIdx.x < s) sdata[threadIdx.x] += sdata[threadIdx.x + s];
        __syncthreads();
    }
    if (threadIdx.x == 0) atomicAdd(accum, sdata[0]);
}

// -------- out = softmax(w/N)[0]*pr + softmax(w/N)[1]*cc --------
__global__ void combine_kernel(const float* __restrict__ pr, const float* __restrict__ cc,
                               const float* __restrict__ wsum, float* __restrict__ out, int n) {
    int i = blockIdx.x * blockDim.x + threadIdx.x;
    if (i >= n) return;
    float w0 = wsum[0] * (1.0f / NN);
    float w1 = wsum[1] * (1.0f / NN);
    float m  = fmaxf(w0, w1);
    float e0 = __expf(w0 - m), e1 = __expf(w1 - m);
    float inv = 1.0f / (e0 + e1);
    out[i] = (e0 * pr[i] + e1 * cc[i]) * inv;
}

static inline int gemm_grid_y(int Nout, int M) {
    int gx  = Nout / 16;
    int gy  = (1024 + gx - 1) / gx;            // target ~1024 blocks
    int cap = (M / 16 + 3) / 4;                // no more than one tile per wave
    return gy < cap ? gy : cap;
}

extern "C" void kernel_launch(void* const* d_in, const int* in_sizes, int n_in,
                              void* d_out, int out_size, void* d_ws, size_t ws_size,
                              hipStream_t stream) {
    (void)in_sizes; (void)n_in; (void)out_size; (void)ws_size;

    const float* k_emb   = (const float*)d_in[0];
    const int*   pr_src  = (const int*)d_in[1];
    const int*   pr_dst  = (const int*)d_in[2];
    const float* pr_ew   = (const float*)d_in[3];
    const int*   cc_src  = (const int*)d_in[4];
    const int*   cc_dst  = (const int*)d_in[5];
    const float* cc_ew   = (const float*)d_in[6];
    const int*   sps_src = (const int*)d_in[7];
    const int*   sps_dst = (const int*)d_in[8];
    const float* pr_Ws0 = (const float*)d_in[9];
    const float* pr_Wn0 = (const float*)d_in[10];
    const float* pr_b0  = (const float*)d_in[11];
    const float* pr_Ws1 = (const float*)d_in[12];
    const float* pr_Wn1 = (const float*)d_in[13];
    const float* pr_b1  = (const float*)d_in[14];
    const float* cc_Ws0 = (const float*)d_in[15];
    const float* cc_Wn0 = (const float*)d_in[16];
    const float* cc_b0  = (const float*)d_in[17];
    const float* cc_Ws1 = (const float*)d_in[18];
    const float* cc_Wn1 = (const float*)d_in[19];
    const float* cc_b1  = (const float*)d_in[20];
    const float* ap_Ws0 = (const float*)d_in[21];
    const float* ap_Wn0 = (const float*)d_in[22];
    const float* ap_b0  = (const float*)d_in[23];
    const float* ap_Ws1 = (const float*)d_in[24];
    const float* ap_Wn1 = (const float*)d_in[25];
    const float* ap_b1  = (const float*)d_in[26];
    const float* att_W  = (const float*)d_in[27];
    const float* att_b  = (const float*)d_in[28];

    // ---- workspace layout (floats) ----
    float* ws = (float*)d_ws;
    size_t off = 0;
    auto alloc = [&](size_t nf) { float* p = ws + off; off += nf; return p; };
    float* deg_pr  = alloc(NN);
    float* deg_cc  = alloc(NN);
    float* deg_sps = alloc(NN);
    float* agg     = alloc((size_t)NN * HH);   // reused: layer0 agg, layer1 agg, proj
    float* hbuf    = alloc((size_t)NN * HH);   // hidden activations
    float* outPR   = alloc((size_t)NN * DD);
    float* outCC   = alloc((size_t)NN * DD);
    float* outAP   = alloc((size_t)NN * DD);
    float* wsum    = alloc(2);
    float* proj    = agg;                      // free after the last branch

    // ---- degrees (zero + count) ----
    hipMemsetAsync(deg_pr, 0, 3ull * NN * sizeof(float), stream);  // pr, cc, sps contiguous
    int degBlocks = (EE + 255) / 256;
    deg_kernel<<<degBlocks, 256, 0, stream>>>(pr_dst,  deg_pr,  EE);
    deg_kernel<<<degBlocks, 256, 0, stream>>>(cc_dst,  deg_cc,  EE);
    deg_kernel<<<degBlocks, 256, 0, stream>>>(sps_dst, deg_sps, EE);

    auto run_branch = [&](const int* src, const int* dstv, const float* ewv,
                          const float* Ws0, const float* Wn0, const float* b0,
                          const float* Ws1, const float* Wn1, const float* b1,
                          const float* deg, int Hmid, float* outb) {
        // ---- layer 0: K=DD -> Hmid, dual input, relu ----
        hipMemsetAsync(agg, 0, (size_t)NN * DD * sizeof(float), stream);
        long long t0 = (long long)EE * (DD / 4);
        scatter_kernel<<<(int)((t0 + 255) / 256), 256, 0, stream>>>(k_emb, src, dstv, ewv, agg, EE, DD);
        normalize_kernel<<<(NN * DD + 255) / 256, 256, 0, stream>>>(agg, deg, NN, DD);
        {
            dim3 grid(Hmid / 16, gemm_grid_y(Hmid, NN));
            size_t sh = 2ull * DD * 16 * sizeof(float);
            gemm_panel_wmma<true, 1><<<grid, dim3(32, 4), sh, stream>>>(
                k_emb, Ws0, agg, Wn0, b0, hbuf, NN, DD, Hmid);
        }
        // ---- layer 1: K=Hmid -> DD, dual input, no act ----
        hipMemsetAsync(agg, 0, (size_t)NN * Hmid * sizeof(float), stream);
        long long t1 = (long long)EE * (Hmid / 4);
        scatter_kernel<<<(int)((t1 + 255) / 256), 256, 0, stream>>>(hbuf, src, dstv, ewv, agg, EE, Hmid);
        normalize_kernel<<<(NN * Hmid + 255) / 256, 256, 0, stream>>>(agg, deg, NN, Hmid);
        {
            dim3 grid(DD / 16, gemm_grid_y(DD, NN));
            size_t sh = 2ull * Hmid * 16 * sizeof(float);
            gemm_panel_wmma<true, 0><<<grid, dim3(32, 4), sh, stream>>>(
                hbuf, Ws1, agg, Wn1, b1, outb, NN, Hmid, DD);
        }
    };

    run_branch(pr_src,  pr_dst,  pr_ew,   pr_Ws0, pr_Wn0, pr_b0, pr_Ws1, pr_Wn1, pr_b1, deg_pr,  HH, outPR);
    run_branch(cc_src,  cc_dst,  cc_ew,   cc_Ws0, cc_Wn0, cc_b0, cc_Ws1, cc_Wn1, cc_b1, deg_cc,  HH, outCC);
    run_branch(sps_src, sps_dst, nullptr, ap_Ws0, ap_Wn0, ap_b0, ap_Ws1, ap_Wn1, ap_b1, deg_sps, DD, outAP);

    // ---- attention fusion ----
    hipMemsetAsync(wsum, 0, 2 * sizeof(float), stream);
    {
        dim3 grid(DD / 16, gemm_grid_y(DD, NN));
        size_t sh = (size_t)DD * 16 * sizeof(float);
        gemm_panel_wmma<false, 2><<<grid, dim3(32, 4), sh, stream>>>(
            outPR, att_W, nullptr, nullptr, att_b, proj, NN, DD, DD);
        dot_reduce_kernel<<<1024, 256, 0, stream>>>(outAP, proj, wsum + 0, NN * DD);
        gemm_panel_wmma<false, 2><<<grid, dim3(32, 4), sh, stream>>>(
            outCC, att_W, nullptr, nullptr, att_b, proj, NN, DD, DD);
        dot_reduce_kernel<<<1024, 256, 0, stream>>>(outAP, proj, wsum + 1, NN * DD);
    }

    combine_kernel<<<(NN * DD + 255) / 256, 256, 0, stream>>>(outPR, outCC, wsum,
                                                              (float*)d_out, NN * DD);
}